// Entmax15Activation_25271587570065
// MI455X (gfx1250) — compile-verified
//
#include <hip/hip_runtime.h>
#include <stdint.h>

// entmax-1.5 over rows of [8192 x 4096] fp32 with bool mask.
// Memory floor: 128MiB in + 32MiB mask + 128MiB out ~= 288MiB -> ~12.4us @ 23.3TB/s.
// Algorithm: bisection on tau in (-1, 0) (f(tau)=sum clip(z-tau)^2 is strictly
// decreasing, f(-1)>=1, f(0)=0 after max-shift). Elements with z <= -1 are
// provably outside the support for every tau probed, so we compact only
// candidates (z > -1, ~1-3% of a row for Gaussian data) into LDS once and run
// the 30 bisection reductions over that tiny set. Compaction uses a
// deterministic prefix scan (no atomics) so output bits are replay-stable.
// Data path: async global->LDS b128 copies (ASYNCcnt) + nontemporal stores.

#define COLS   4096
#define TPB    256
#define CHUNKS 4            // 4 float4 chunks/thread * 256 threads = 4096 floats
#define NITER  30
#define NEG_HALF -5000.0f   // NEG_FILL(-1e4) / 2

typedef __attribute__((ext_vector_type(4))) float v4f;

__device__ __forceinline__ void async_load_b128(uint32_t lds_addr, uint32_t voff,
                                                const void* sbase) {
  // GLOBAL_LOAD_ASYNC_TO_LDS_B128: vdst = LDS byte address, vaddr = u32 byte
  // offset, saddr = 64-bit base (GVS mode). Tracked by ASYNCcnt.
  asm volatile("global_load_async_to_lds_b128 %0, %1, %2 offset:0"
               :: "v"(lds_addr), "v"(voff), "s"(sbase) : "memory");
}

__global__ __launch_bounds__(TPB) void entmax15_rows_kernel(
    const float* __restrict__ scores,
    const unsigned char* __restrict__ mask,
    float* __restrict__ out, int rows)
{
  __shared__ __align__(16) float          s_scores[COLS];   // 16 KB
  __shared__ __align__(16) unsigned char  s_mask[COLS];     //  4 KB
  __shared__ __align__(16) float          s_cand[COLS];     // 16 KB (worst case)
  __shared__ float s_red[16];   // 8 wave slots x 2 (parity ping-pong)
  __shared__ int   s_wsum[8];   // per-wave candidate counts

  const int row = blockIdx.x;
  if (row >= rows) return;
  const int tid  = threadIdx.x;
  const int lane = tid & 31;
  const int wid  = tid >> 5;

  const float*         srow = scores + (size_t)row * COLS;
  const unsigned char* mrow = mask   + (size_t)row * COLS;

  // Low 32 bits of the generic pointer to an LDS object are the wave-relative
  // LDS byte address (flat->LDS aperture truncation, ISA 10.2).
  const uint32_t lds_s = (uint32_t)(size_t)&s_scores[0];
  const uint32_t lds_m = (uint32_t)(size_t)&s_mask[0];

  // ---- stage row into LDS with async copies (4x16B scores + 1x16B mask) ----
#pragma unroll
  for (int j = 0; j < CHUNKS; ++j) {
    uint32_t off = (uint32_t)(tid + j * TPB) * 16u;
    async_load_b128(lds_s + off, off, (const void*)srow);
  }
  {
    uint32_t off = (uint32_t)tid * 16u;          // 256*16B = 4096 mask bytes
    async_load_b128(lds_m + off, off, (const void*)mrow);
  }
  asm volatile("s_wait_asynccnt 0" ::: "memory");
  __syncthreads();

  // ---- build z = mask ? score/2 : -5000 in registers; local max ----
  v4f  z[CHUNKS];
  float lmax = NEG_HALF;
#pragma unroll
  for (int j = 0; j < CHUNKS; ++j) {
    const int c4 = tid + j * TPB;
    v4f s       = *(const v4f*)&s_scores[4 * c4];
    uint32_t mw = *(const uint32_t*)&s_mask[4 * c4];
    v4f zz;
    zz.x = (mw & 0x000000ffu) ? 0.5f * s.x : NEG_HALF;
    zz.y = (mw & 0x0000ff00u) ? 0.5f * s.y : NEG_HALF;
    zz.z = (mw & 0x00ff0000u) ? 0.5f * s.z : NEG_HALF;
    zz.w = (mw & 0xff000000u) ? 0.5f * s.w : NEG_HALF;
    z[j] = zz;
    lmax = fmaxf(lmax, fmaxf(fmaxf(zz.x, zz.y), fmaxf(zz.z, zz.w)));
  }

  // ---- block max (wave shuffle tree + 8-slot LDS) ----
  lmax = fmaxf(lmax, __shfl_xor(lmax, 16, 32));
  lmax = fmaxf(lmax, __shfl_xor(lmax,  8, 32));
  lmax = fmaxf(lmax, __shfl_xor(lmax,  4, 32));
  lmax = fmaxf(lmax, __shfl_xor(lmax,  2, 32));
  lmax = fmaxf(lmax, __shfl_xor(lmax,  1, 32));
  if (lane == 0) s_red[wid] = lmax;
  __syncthreads();
  float rmax = s_red[0];
#pragma unroll
  for (int k = 1; k < 8; ++k) rmax = fmaxf(rmax, s_red[k]);

  // ---- shift by row max; count candidates (z > -1) ----
  int cnt = 0;
#pragma unroll
  for (int j = 0; j < CHUNKS; ++j) {
    z[j].x -= rmax; z[j].y -= rmax; z[j].z -= rmax; z[j].w -= rmax;
    cnt += (z[j].x > -1.0f) + (z[j].y > -1.0f) + (z[j].z > -1.0f) + (z[j].w > -1.0f);
  }

  // ---- deterministic block-wide exclusive scan of counts ----
  int scan = cnt;                          // wave inclusive scan
#pragma unroll
  for (int off = 1; off < 32; off <<= 1) {
    int n = __shfl_up(scan, off, 32);
    if (lane >= off) scan += n;
  }
  if (lane == 31) s_wsum[wid] = scan;      // per-wave totals
  __syncthreads();
  int wbase = 0;
  for (int k = 0; k < wid; ++k) wbase += s_wsum[k];
  int ncand = 0;
#pragma unroll
  for (int k = 0; k < 8; ++k) ncand += s_wsum[k];
  int pos = wbase + (scan - cnt);          // exclusive prefix for this thread

  // ---- compact candidates into LDS in a fixed, deterministic order ----
#pragma unroll
  for (int j = 0; j < CHUNKS; ++j) {
    if (z[j].x > -1.0f) s_cand[pos++] = z[j].x;
    if (z[j].y > -1.0f) s_cand[pos++] = z[j].y;
    if (z[j].z > -1.0f) s_cand[pos++] = z[j].z;
    if (z[j].w > -1.0f) s_cand[pos++] = z[j].w;
  }
  __syncthreads();   // also separates s_red reads above from bisection writes

  // ---- bisection on tau in (-1, 0): f(tau) = sum clip(z - tau)^2 = 1 ----
  float lo = -1.0f, hi = 0.0f;
  for (int it = 0; it < NITER; ++it) {
    const float tau = 0.5f * (lo + hi);
    float acc = 0.0f;
    for (int i = tid; i < ncand; i += TPB) {
      float d = s_cand[i] - tau;
      d = fmaxf(d, 0.0f);
      acc = fmaf(d, d, acc);
    }
    acc += __shfl_xor(acc, 16, 32);
    acc += __shfl_xor(acc,  8, 32);
    acc += __shfl_xor(acc,  4, 32);
    acc += __shfl_xor(acc,  2, 32);
    acc += __shfl_xor(acc,  1, 32);
    const int base = (it & 1) << 3;        // parity slots -> 1 barrier/iter
    if (lane == 0) s_red[base + wid] = acc;
    __syncthreads();
    float total = s_red[base];
#pragma unroll
    for (int k = 1; k < 8; ++k) total += s_red[base + k];
    if (total >= 1.0f) lo = tau; else hi = tau;   // f decreasing
  }
  const float tau = 0.5f * (lo + hi);

  // ---- p = clip(z - tau)^2 ; nontemporal 128-bit stores ----
  float* orow = out + (size_t)row * COLS;
#pragma unroll
  for (int j = 0; j < CHUNKS; ++j) {
    const int c4 = tid + j * TPB;
    v4f p; float d;
    d = fmaxf(z[j].x - tau, 0.0f); p.x = d * d;
    d = fmaxf(z[j].y - tau, 0.0f); p.y = d * d;
    d = fmaxf(z[j].z - tau, 0.0f); p.z = d * d;
    d = fmaxf(z[j].w - tau, 0.0f); p.w = d * d;
    __builtin_nontemporal_store(p, (v4f*)&orow[4 * c4]);
  }
}

extern "C" void kernel_launch(void* const* d_in, const int* in_sizes, int n_in,
                              void* d_out, int out_size, void* d_ws, size_t ws_size,
                              hipStream_t stream) {
  const float*         scores = (const float*)d_in[0];
  const unsigned char* mask   = (const unsigned char*)d_in[1]; // np.bool_ = 1 byte
  float*               out    = (float*)d_out;
  const int rows = in_sizes[0] / COLS;     // 8192
  entmax15_rows_kernel<<<rows, TPB, 0, stream>>>(scores, mask, out, rows);
}